// CELoss_58042188038587
// MI455X (gfx1250) — compile-verified
//
#include <hip/hip_runtime.h>

// Problem constants from the reference: B=8, C=19, H=W=512.
#define NCLASS 19
#define HW     (512 * 512)          // 2^18 elements per channel plane
#define HW4    (HW / 4)             // channel plane in float4 units (2^16)
#define NVOX   (8 * 512 * 512)      // B*H*W = 2,097,152 voxels
#define CHW    (NCLASS * HW)
#define CHW4   (CHW / 4)

struct alignas(16) F4 { float v[4]; };   // lowers to global_load_b128

// Workspace layout (in d_ws): float sums[19]; unsigned counts[19]
__global__ void ce_zero_ws(float* __restrict__ sums, unsigned* __restrict__ cnts) {
    int t = threadIdx.x;
    if (t < NCLASS) { sums[t] = 0.0f; cnts[t] = 0u; }
}

__global__ __launch_bounds__(256) void ce_pass(const F4* __restrict__ predict,
                                               const F4* __restrict__ target,
                                               float* __restrict__ sums,
                                               unsigned* __restrict__ cnts) {
    __shared__ float    s_sum[NCLASS];
    __shared__ unsigned s_cnt[NCLASS];
    const int t = threadIdx.x;
    if (t < NCLASS) { s_sum[t] = 0.0f; s_cnt[t] = 0u; }
    __syncthreads();

    // One thread = 4 consecutive voxels (one float4). Grid sized exactly.
    const unsigned q    = blockIdx.x * 256u + t;       // float4-voxel id < NVOX/4
    const unsigned b    = q >> 16;                     // (4q) / HW
    const unsigned hw4  = q & (HW4 - 1u);              // (4q mod HW) / 4
    const unsigned base = b * CHW4 + hw4;              // float4 index

    // Prefetch the strided predict stream (gfx1250 global_prefetch_b8)
    // while the dependent target/argmax chain executes.
#pragma unroll
    for (int c = 0; c < NCLASS; ++c)
        __builtin_prefetch(predict + base + c * HW4, 0, 0);

    // argmax over target channels; first-max wins (matches jnp.argmax).
    int   mask[4];
    float tmax[4];
    {
        F4 t0 = target[base];
#pragma unroll
        for (int j = 0; j < 4; ++j) { mask[j] = 0; tmax[j] = t0.v[j]; }
    }
#pragma unroll
    for (int c = 1; c < NCLASS; ++c) {
        F4 tc = target[base + c * HW4];
#pragma unroll
        for (int j = 0; j < 4; ++j)
            if (tc.v[j] > tmax[j]) { tmax[j] = tc.v[j]; mask[j] = c; }
    }

    // predict channels into registers (static unroll -> VGPRs, no scratch)
    float p[NCLASS][4];
    float m[4] = {-INFINITY, -INFINITY, -INFINITY, -INFINITY};
#pragma unroll
    for (int c = 0; c < NCLASS; ++c) {
        F4 pc = predict[base + c * HW4];
#pragma unroll
        for (int j = 0; j < 4; ++j) {
            p[c][j] = pc.v[j];
            m[j] = fmaxf(m[j], pc.v[j]);
        }
    }

    // 4 independent logsumexp chains keep the v_exp_f32 pipeline busy.
    float s[4] = {0.f, 0.f, 0.f, 0.f};
    float pm[4];
#pragma unroll
    for (int j = 0; j < 4; ++j) pm[j] = p[0][j];
#pragma unroll
    for (int c = 0; c < NCLASS; ++c) {
#pragma unroll
        for (int j = 0; j < 4; ++j) {
            s[j] += __expf(p[c][j] - m[j]);
            pm[j] = (c == mask[j]) ? p[c][j] : pm[j];  // static-index select
        }
    }

#pragma unroll
    for (int j = 0; j < 4; ++j) {
        const float ce = (m[j] + __logf(s[j])) - pm[j];   // -log_softmax at mask
        atomicAdd(&s_sum[mask[j]], ce);                   // ds_add_f32
        atomicAdd(&s_cnt[mask[j]], 1u);                   // ds_add_u32
    }
    __syncthreads();

    if (t < NCLASS && s_cnt[t] != 0u) {
        atomicAdd(&sums[t], s_sum[t]);                    // global_atomic_add_f32
        atomicAdd(&cnts[t], s_cnt[t]);                    // global_atomic_add_u32
    }
}

__global__ void ce_finalize(const float* __restrict__ sums,
                            const unsigned* __restrict__ cnts,
                            float* __restrict__ out) {
    if (threadIdx.x == 0 && blockIdx.x == 0) {
        const float Nv = (float)NVOX;
        float total = 0.0f;
        for (int c = 0; c < NCLASS; ++c) {
            float cnt = (float)cnts[c];
            float w   = (cnt > 0.0f) ? __logf(Nv / cnt) : 1.0f;  // absent class -> 1.0
            total += sums[c] * w;
        }
        out[0] = total / Nv;
    }
}

extern "C" void kernel_launch(void* const* d_in, const int* in_sizes, int n_in,
                              void* d_out, int out_size, void* d_ws, size_t ws_size,
                              hipStream_t stream) {
    (void)in_sizes; (void)n_in; (void)out_size; (void)ws_size;
    const F4* predict = (const F4*)d_in[0];
    const F4* target  = (const F4*)d_in[1];
    float*    out  = (float*)d_out;
    float*    sums = (float*)d_ws;
    unsigned* cnts = (unsigned*)((char*)d_ws + NCLASS * sizeof(float));

    ce_zero_ws<<<1, 32, 0, stream>>>(sums, cnts);
    ce_pass<<<(NVOX / 4) / 256, 256, 0, stream>>>(predict, target, sums, cnts);
    ce_finalize<<<1, 32, 0, stream>>>(sums, cnts, out);
}